// AutoregressiveNetwork_72919954751764
// MI455X (gfx1250) — compile-verified
//
#include <hip/hip_runtime.h>

typedef __attribute__((ext_vector_type(2))) float v2f;
typedef __attribute__((ext_vector_type(8))) float v8f;

#define BATCH 16384
#define DIM   64
#define HID   64
#define WSTR  68                 // padded LDS row stride (floats): conflict-free A/B access
#define ROWS_PER_BLOCK 128       // 8 waves x 16 rows
#define OUT_OFF ((size_t)BATCH * DIM)

// bias + relu on a 16x16 f32 D-tile, store to per-wave LDS h-buffer
__device__ __forceinline__ void relu_store_tile(float* myH, const v8f& acc, float bias,
                                                int colBase, int hi, int l15) {
#pragma unroll
  for (int v = 0; v < 8; ++v) {
    float val = acc[v] + bias;
    val = fmaxf(val, 0.0f);
    myH[(v + 8 * hi) * WSTR + colBase + l15] = val;
  }
}

__global__ __launch_bounds__(256) void ar_main_kernel(
    const float* __restrict__ x,
    const float* __restrict__ W1, const float* __restrict__ B1,
    const float* __restrict__ W2, const float* __restrict__ B2,
    const float* __restrict__ W3, const float* __restrict__ B3,
    float* __restrict__ out)
{
  __shared__ __align__(16) float ldsW[HID * WSTR];        // transposed weights Wt[h][j]
  __shared__ __align__(16) float ldsH[8 * 16 * WSTR];     // per-wave 16x64 activation tile

  const int tid  = threadIdx.x;
  const int wave = tid >> 5;
  const int lane = tid & 31;
  const int l15  = lane & 15;
  const int hi   = lane >> 4;            // 0 or 1 (wave half)
  const int net1 = blockIdx.y;           // 0..62  -> network n = net1+1
  const int n    = net1 + 1;
  const int r0   = blockIdx.x * ROWS_PER_BLOCK + wave * 16;

  float* myH = &ldsH[wave * 16 * WSTR];

  // ---- stage masked W1 (transposed) into LDS ----
  {
    const float* src = W1 + (size_t)net1 * (DIM * HID);
    for (int e = tid; e < DIM * HID; e += 256) {
      int j = e >> 6;          // input dim (K)
      int h = e & 63;          // hidden (N)
      float w = src[e];
      if (j >= n) w = 0.0f;    // autoregressive mask: network n sees x[:, :n]
      ldsW[h * WSTR + j] = w;
    }
  }
  __syncthreads();

  // ---- layer 1: h1 = relu(x @ W1m + B1) ----
  v8f acc0 = {}, acc1 = {}, acc2 = {}, acc3 = {};
  {
    const float* xrow = x + (size_t)(r0 + l15) * DIM;
    for (int kc = 0; kc < 16; ++kc) {
      int kb = kc * 4 + hi * 2;
      v2f a  = *(const v2f*)(xrow + kb);
      v2f b0 = *(const v2f*)&ldsW[(0 * 16 + l15) * WSTR + kb];
      v2f b1 = *(const v2f*)&ldsW[(1 * 16 + l15) * WSTR + kb];
      v2f b2 = *(const v2f*)&ldsW[(2 * 16 + l15) * WSTR + kb];
      v2f b3 = *(const v2f*)&ldsW[(3 * 16 + l15) * WSTR + kb];
      acc0 = __builtin_amdgcn_wmma_f32_16x16x4_f32(false, a, false, b0, (short)0, acc0, false, false);
      acc1 = __builtin_amdgcn_wmma_f32_16x16x4_f32(false, a, false, b1, (short)0, acc1, false, false);
      acc2 = __builtin_amdgcn_wmma_f32_16x16x4_f32(false, a, false, b2, (short)0, acc2, false, false);
      acc3 = __builtin_amdgcn_wmma_f32_16x16x4_f32(false, a, false, b3, (short)0, acc3, false, false);
    }
  }
  {
    const float* bp = B1 + net1 * HID;
    relu_store_tile(myH, acc0, bp[ 0 + l15],  0, hi, l15);
    relu_store_tile(myH, acc1, bp[16 + l15], 16, hi, l15);
    relu_store_tile(myH, acc2, bp[32 + l15], 32, hi, l15);
    relu_store_tile(myH, acc3, bp[48 + l15], 48, hi, l15);
  }
  __syncthreads();   // all waves done reading W1 from ldsW; h1 tiles stored

  // ---- stage W2 (transposed, no mask) into LDS ----
  {
    const float* src = W2 + (size_t)net1 * (DIM * HID);
    for (int e = tid; e < DIM * HID; e += 256) {
      int j = e >> 6;
      int h = e & 63;
      ldsW[h * WSTR + j] = src[e];
    }
  }
  __syncthreads();

  // ---- layer 2: h2 = relu(h1 @ W2 + B2) ----
  v8f d0 = {}, d1 = {}, d2 = {}, d3 = {};
  for (int kc = 0; kc < 16; ++kc) {
    int kb = kc * 4 + hi * 2;
    v2f a  = *(const v2f*)&myH[l15 * WSTR + kb];
    v2f b0 = *(const v2f*)&ldsW[(0 * 16 + l15) * WSTR + kb];
    v2f b1 = *(const v2f*)&ldsW[(1 * 16 + l15) * WSTR + kb];
    v2f b2 = *(const v2f*)&ldsW[(2 * 16 + l15) * WSTR + kb];
    v2f b3 = *(const v2f*)&ldsW[(3 * 16 + l15) * WSTR + kb];
    d0 = __builtin_amdgcn_wmma_f32_16x16x4_f32(false, a, false, b0, (short)0, d0, false, false);
    d1 = __builtin_amdgcn_wmma_f32_16x16x4_f32(false, a, false, b1, (short)0, d1, false, false);
    d2 = __builtin_amdgcn_wmma_f32_16x16x4_f32(false, a, false, b2, (short)0, d2, false, false);
    d3 = __builtin_amdgcn_wmma_f32_16x16x4_f32(false, a, false, b3, (short)0, d3, false, false);
  }
  __syncthreads();   // all h1 reads complete before overwriting with h2
  {
    const float* bp = B2 + net1 * HID;
    relu_store_tile(myH, d0, bp[ 0 + l15],  0, hi, l15);
    relu_store_tile(myH, d1, bp[16 + l15], 16, hi, l15);
    relu_store_tile(myH, d2, bp[32 + l15], 32, hi, l15);
    relu_store_tile(myH, d3, bp[48 + l15], 48, hi, l15);
  }
  __syncthreads();

  // ---- layer 3: out = h2 @ W3 + B3  (64x2, VALU) ----
  {
    const int o = hi;                       // 0 = scale, 1 = translation
    float accv = B3[net1 * 2 + o];
    const float* w3 = W3 + (size_t)net1 * (HID * 2) + o;
    const float* hrow = &myH[l15 * WSTR];
#pragma unroll 8
    for (int c = 0; c < HID; ++c) {
      accv += hrow[c] * w3[c * 2];
    }
    const size_t b = (size_t)(r0 + l15);
    if (o == 0) {
      float s = fminf(5.0f, fmaxf(-5.0f, accv));
      out[b * DIM + n] = s;
    } else {
      out[OUT_OFF + b * DIM + n] = accv;
    }
  }
}

// network 0: constant output broadcast across the batch
__global__ __launch_bounds__(256) void ar_net0_kernel(
    const float* __restrict__ w0, const float* __restrict__ b0,
    const float* __restrict__ v0, const float* __restrict__ c0,
    float* __restrict__ out)
{
  const int b = blockIdx.x * blockDim.x + threadIdx.x;   // 0..BATCH-1
  float s = c0[0];
  float t = c0[1];
#pragma unroll 8
  for (int h = 0; h < HID; ++h) {
    float hv = fmaxf(w0[h] + b0[h], 0.0f);
    s += hv * v0[h * 2 + 0];
    t += hv * v0[h * 2 + 1];
  }
  s = fminf(5.0f, fmaxf(-5.0f, s));
  out[(size_t)b * DIM + 0] = s;
  out[OUT_OFF + (size_t)b * DIM + 0] = t;
}

extern "C" void kernel_launch(void* const* d_in, const int* in_sizes, int n_in,
                              void* d_out, int out_size, void* d_ws, size_t ws_size,
                              hipStream_t stream) {
  const float* x  = (const float*)d_in[0];
  const float* w0 = (const float*)d_in[1];
  const float* b0 = (const float*)d_in[2];
  const float* v0 = (const float*)d_in[3];
  const float* c0 = (const float*)d_in[4];
  const float* W1 = (const float*)d_in[5];
  const float* B1 = (const float*)d_in[6];
  const float* W2 = (const float*)d_in[7];
  const float* B2 = (const float*)d_in[8];
  const float* W3 = (const float*)d_in[9];
  const float* B3 = (const float*)d_in[10];
  float* out = (float*)d_out;

  dim3 grid(BATCH / ROWS_PER_BLOCK, DIM - 1, 1);
  ar_main_kernel<<<grid, 256, 0, stream>>>(x, W1, B1, W2, B2, W3, B3, out);
  ar_net0_kernel<<<BATCH / 256, 256, 0, stream>>>(w0, b0, v0, c0, out);
}